// GNNBlock_12695923327377
// MI455X (gfx1250) — compile-verified
//
#include <hip/hip_runtime.h>
#include <math.h>

#define N_NODES 50000
#define N_EDGES 600000
#define D 128
#define EPSV 1e-5f

typedef __attribute__((ext_vector_type(2))) float v2f;
typedef __attribute__((ext_vector_type(8))) float v8f;

// ---------------------------------------------------------------- fill
__global__ void fill_f32(float* __restrict__ p, float v, int n) {
    int i = blockIdx.x * blockDim.x + threadIdx.x;
    if (i < n) p[i] = v;
}

// ---------------------------------------------------------------- degree (in-degree over col, self loop pre-seeded as 1.0)
__global__ void deg_kernel(const int* __restrict__ col, float* __restrict__ deg, int E) {
    int e = blockIdx.x * blockDim.x + threadIdx.x;
    if (e < E) atomicAdd(&deg[col[e]], 1.0f);
}

__global__ void dinv_kernel(const float* __restrict__ deg, float* __restrict__ dinv, int n) {
    int i = blockIdx.x * blockDim.x + threadIdx.x;
    if (i < n) {
        float d = deg[i];
        dinv[i] = d > 0.0f ? rsqrtf(d) : 0.0f;
    }
}

// ---------------------------------------------------------------- GEMM: xw = x @ W^T  via V_WMMA_F32_16X16X4_F32
// block = 256 threads = 8 waves; wave w computes the 16x16 tile at
// rows [16*blockIdx.x, +16), cols [16*w, +16).  D_OUT = 128 = 8 tiles.
__global__ void __launch_bounds__(256) gemm_xwT(const float* __restrict__ x,
                                                const float* __restrict__ W,
                                                float* __restrict__ xw) {
    const int lane = threadIdx.x & 31;
    const int wave = threadIdx.x >> 5;
    const int m0 = blockIdx.x << 4;
    const int n0 = wave << 4;
    const int r    = lane & 15;        // row within tile (A: M, B: N)
    const int koff = (lane >> 4) << 1; // lanes 16..31 hold K+2,K+3

    const float* aRow = x + (size_t)(m0 + r) * D + koff; // A[M=r, K..]
    const float* bRow = W + (size_t)(n0 + r) * D + koff; // B[K, N=r] = W[N=r, K]

    v8f c = {};
#pragma unroll
    for (int k = 0; k < D; k += 4) {
        v2f a = *(const v2f*)(aRow + k);
        v2f b = *(const v2f*)(bRow + k);
        c = __builtin_amdgcn_wmma_f32_16x16x4_f32(false, a, false, b,
                                                  (short)0, c, false, false);
    }

    // C/D layout: VGPR i -> M = m0 + i (+8 for lanes 16..31), N = n0 + (lane&15)
    const int mBase = m0 + ((lane >> 4) << 3);
    const int n = n0 + r;
#pragma unroll
    for (int i = 0; i < 8; ++i)
        xw[(size_t)(mBase + i) * D + n] = c[i];
}

// ---------------------------------------------------------------- acc init: self-loop term  acc[n] = xw[n] * dinv[n]^2
__global__ void acc_init(const float* __restrict__ xw, const float* __restrict__ dinv,
                         float* __restrict__ acc, int total) {
    int i = blockIdx.x * blockDim.x + threadIdx.x;
    if (i < total) {
        int node = i >> 7; // / D
        float di = dinv[node];
        acc[i] = xw[i] * di * di;
    }
}

// ---------------------------------------------------------------- edge scatter: 32 lanes per edge, float4 per lane
__global__ void __launch_bounds__(256) scatter_kernel(const int* __restrict__ rows,
                                                      const int* __restrict__ cols,
                                                      const float* __restrict__ dinv,
                                                      const float* __restrict__ xw,
                                                      float* __restrict__ acc, int E) {
    int t = blockIdx.x * blockDim.x + threadIdx.x;
    int e = t >> 5;
    if (e >= E) return;
    int lane = t & 31;
    int r = rows[e];
    int c = cols[e];
    float norm = dinv[r] * dinv[c];
    const float4 v = *(const float4*)(xw + (size_t)r * D + (lane << 2));
    float* dst = acc + (size_t)c * D + (lane << 2);
    atomicAdd(dst + 0, v.x * norm);
    atomicAdd(dst + 1, v.y * norm);
    atomicAdd(dst + 2, v.z * norm);
    atomicAdd(dst + 3, v.w * norm);
}

// ---------------------------------------------------------------- PairNorm reductions: colsum[f] and sum(out^2)
#define NODES_PER_BLOCK 256
__global__ void __launch_bounds__(128) colsum_kernel(const float* __restrict__ acc,
                                                     float* __restrict__ colsum,
                                                     float* __restrict__ sumsq, int N) {
    __shared__ float sh[128];
    int f = threadIdx.x;
    int n0 = blockIdx.x * NODES_PER_BLOCK;
    int n1 = n0 + NODES_PER_BLOCK; if (n1 > N) n1 = N;
    float s = 0.0f, sq = 0.0f;
    for (int n = n0; n < n1; ++n) {
        float v = acc[(size_t)n * D + f];
        s += v;
        sq += v * v;
    }
    atomicAdd(&colsum[f], s);
    sh[f] = sq;
    __syncthreads();
    for (int stride = 64; stride > 0; stride >>= 1) {
        if (f < stride) sh[f] += sh[f + stride];
        __syncthreads();
    }
    if (f == 0) atomicAdd(sumsq, sh[0]);
}

// ---------------------------------------------------------------- stats: mean[f], invden  (sum xc^2 = sumsq - N*sum mean^2)
__global__ void __launch_bounds__(128) stats_kernel(const float* __restrict__ colsum,
                                                    const float* __restrict__ sumsq,
                                                    float* __restrict__ mean,
                                                    float* __restrict__ invden, int N) {
    __shared__ float sh[128];
    int f = threadIdx.x;
    float cs = colsum[f];
    float m = cs / (float)N;
    mean[f] = m;
    sh[f] = cs * m;           // = colsum^2 / N
    __syncthreads();
    for (int stride = 64; stride > 0; stride >>= 1) {
        if (f < stride) sh[f] += sh[f + stride];
        __syncthreads();
    }
    if (f == 0) {
        float ss = sumsq[0] - sh[0];          // sum over all of xc^2
        float den = sqrtf(EPSV + ss / (float)N);
        invden[0] = 1.0f / den;
    }
}

// ---------------------------------------------------------------- finalize: y = (out-mean)*invden; out = relu(y)+y
__global__ void final_kernel(float* __restrict__ out, const float* __restrict__ mean,
                             const float* __restrict__ invden, int total) {
    int i = blockIdx.x * blockDim.x + threadIdx.x;
    if (i < total) {
        float y = (out[i] - mean[i & (D - 1)]) * invden[0];
        out[i] = y > 0.0f ? 2.0f * y : y;
    }
}

extern "C" void kernel_launch(void* const* d_in, const int* in_sizes, int n_in,
                              void* d_out, int out_size, void* d_ws, size_t ws_size,
                              hipStream_t stream) {
    const float* x  = (const float*)d_in[0];            // [N, 128]
    const float* W  = (const float*)d_in[1];            // [128, 128]
    const int*   ei = (const int*)d_in[2];              // [2, E] flat
    const int* rows = ei;                               // edge_index[0] (sources)
    const int* cols = ei + N_EDGES;                     // edge_index[1] (targets)
    float* out = (float*)d_out;                         // [N, 128] — used as accumulator

    // workspace layout (floats)
    float* ws     = (float*)d_ws;
    float* xw     = ws;                      // 6,400,000
    float* deg    = xw + (size_t)N_NODES * D;  // 50,000
    float* dinv   = deg + N_NODES;           // 50,000
    float* colsum = dinv + N_NODES;          // 128
    float* sumsq  = colsum + D;              // 1
    float* mean   = sumsq + 1;               // 128
    float* invden = mean + D;                // 1

    const int TB = 256;
    const int total = N_NODES * D;

    // 1. init: deg = 1.0 (self loops), colsum/sumsq = 0
    fill_f32<<<(N_NODES + TB - 1) / TB, TB, 0, stream>>>(deg, 1.0f, N_NODES);
    fill_f32<<<1, TB, 0, stream>>>(colsum, 0.0f, D + 1);

    // 2. degree over targets, then dinv = rsqrt(deg)
    deg_kernel<<<(N_EDGES + TB - 1) / TB, TB, 0, stream>>>(cols, deg, N_EDGES);
    dinv_kernel<<<(N_NODES + TB - 1) / TB, TB, 0, stream>>>(deg, dinv, N_NODES);

    // 3. xw = x @ W^T  (WMMA fp32): 3125 blocks x 8 waves, one 16x16 tile/wave
    gemm_xwT<<<N_NODES / 16, 256, 0, stream>>>(x, W, xw);

    // 4. acc = self-loop contribution
    acc_init<<<(total + TB - 1) / TB, TB, 0, stream>>>(xw, dinv, out, total);

    // 5. edge scatter-add (32 lanes/edge)
    {
        long long threads = (long long)N_EDGES * 32;
        int blocks = (int)((threads + TB - 1) / TB);
        scatter_kernel<<<blocks, TB, 0, stream>>>(rows, cols, dinv, xw, out, N_EDGES);
    }

    // 6. PairNorm reductions + stats
    colsum_kernel<<<(N_NODES + NODES_PER_BLOCK - 1) / NODES_PER_BLOCK, 128, 0, stream>>>(
        out, colsum, sumsq, N_NODES);
    stats_kernel<<<1, 128, 0, stream>>>(colsum, sumsq, mean, invden, N_NODES);

    // 7. finalize in place
    final_kernel<<<(total + TB - 1) / TB, TB, 0, stream>>>(out, mean, invden, total);
}